// LongTermGNN_1262720385615
// MI455X (gfx1250) — compile-verified
//
#include <hip/hip_runtime.h>
#include <math.h>

#define N_ 50000
#define E_ 800000
#define D_ 128
#define R_ 8
#define B_ 4
#define L_ 2
#define EPS_ 1e-5f
#define SLOPE_ 0.2f

typedef float v2f __attribute__((ext_vector_type(2)));
typedef float v8f __attribute__((ext_vector_type(8)));

// ---------------- helpers ----------------
__device__ __forceinline__ float wave_sum32(float v) {
#pragma unroll
  for (int off = 16; off > 0; off >>= 1) v += __shfl_xor(v, off, 32);
  return v;
}

__device__ __forceinline__ float atomic_add_f32(float* p, float v) {
  return __hip_atomic_fetch_add(p, v, __ATOMIC_RELAXED, __HIP_MEMORY_SCOPE_AGENT);
}

// monotone float -> uint key (for atomicMax-based segment max)
__device__ __forceinline__ unsigned f2key(float f) {
  unsigned u = __float_as_uint(f);
  return (u & 0x80000000u) ? ~u : (u | 0x80000000u);
}
__device__ __forceinline__ float key2f(unsigned k) {
  unsigned u = (k & 0x80000000u) ? (k ^ 0x80000000u) : ~k;
  float f = __uint_as_float(u);
  // reference: amax = where(isfinite(amax), amax, 0)
  if ((u & 0x7F800000u) == 0x7F800000u) f = 0.0f;
  return f;
}

// ---------------- kernels ----------------

// w[r,i,o] = sum_b att_r[r,b] * basis[b,i,o]   (R*D*D threads)
__global__ void make_w(const float* __restrict__ att_r,
                       const float* __restrict__ basis,
                       float* __restrict__ w) {
  int idx = blockIdx.x * blockDim.x + threadIdx.x;
  if (idx >= R_ * D_ * D_) return;
  int r = idx / (D_ * D_);
  int io = idx - r * (D_ * D_);
  float acc = 0.f;
#pragma unroll
  for (int b = 0; b < B_; ++b) acc += att_r[r * B_ + b] * basis[b * D_ * D_ + io];
  w[idx] = acc;
}

// 9 GEMMs: out[mat] = x @ W[mat]  (mat 0..R-1 -> xw, mat R -> xr via root)
// grid = (N/16, 1, R+1), block = 256 (8 waves); wave id == column tile (D/16 == 8)
__global__ void gemm_xw(const float* __restrict__ x,
                        const float* __restrict__ w,
                        const float* __restrict__ root,
                        float* __restrict__ xw,
                        float* __restrict__ xr) {
  const int wave = threadIdx.x >> 5;
  const int lane = threadIdx.x & 31;
  const int rowTile = blockIdx.x;
  const int mat = blockIdx.z;
  const float* __restrict__ W = (mat < R_) ? (w + (size_t)mat * D_ * D_) : root;
  float* __restrict__ out = (mat < R_) ? (xw + (size_t)mat * N_ * D_) : xr;

  const int m = lane & 15;              // row within tile (A) / col within tile (B)
  const int kk = (lane >> 4) << 1;      // K sub-offset for this half-wave
  const int n = (wave << 4) + m;        // output column
  const int row = (rowTile << 4) + m;   // A row this lane feeds
  const float* __restrict__ arow = x + (size_t)row * D_;

  v8f acc = {};
#pragma unroll
  for (int k0 = 0; k0 < D_; k0 += 4) {
    v2f a, b;
    // A 16x4 f32 layout: VGPR0 = K {0,2}, VGPR1 = K {1,3} split by half-wave
    a[0] = arow[k0 + kk];
    a[1] = arow[k0 + kk + 1];
    // B 4x16 f32 layout mirrors A on the K dimension
    b[0] = W[(size_t)(k0 + kk) * D_ + n];
    b[1] = W[(size_t)(k0 + kk + 1) * D_ + n];
    acc = __builtin_amdgcn_wmma_f32_16x16x4_f32(
        /*neg_a=*/false, a, /*neg_b=*/false, b,
        /*c_mod=*/(short)0, acc, /*reuse_a=*/false, /*reuse_b=*/false);
  }
  // C/D 16x16 f32 layout: VGPR i -> M = i (+8 for upper half-wave), N = lane&15
  const int mbase = (rowTile << 4) + ((lane >> 4) << 3);
#pragma unroll
  for (int i = 0; i < 8; ++i)
    out[(size_t)(mbase + i) * D_ + n] = acc[i];
}

// zero aggr (N*D), amax keys (N), denom (N)
__global__ void zero_bufs(float* __restrict__ aggr,
                          unsigned* __restrict__ amaxkey,
                          float* __restrict__ denom) {
  int idx = blockIdx.x * blockDim.x + threadIdx.x;
  if (idx < N_ * D_) aggr[idx] = 0.f;
  if (idx < N_) { amaxkey[idx] = 0u; denom[idx] = 0.f; }
}

// one wave32 per edge: alpha = leaky_relu(<x_i,att_l> + <x_j,att_r>); atomicMax segment max
__global__ void edge_alpha(const float* __restrict__ xw,
                           const float* __restrict__ xr,
                           const float* __restrict__ att,
                           const int* __restrict__ src,
                           const int* __restrict__ dst,
                           const int* __restrict__ et,
                           float* __restrict__ alpha,
                           unsigned* __restrict__ amaxkey) {
  int e = (blockIdx.x * blockDim.x + threadIdx.x) >> 5;
  int lane = threadIdx.x & 31;
  if (e >= E_) return;
  int s = src[e], d = dst[e], r = et[e];
  const float4* xi = (const float4*)(xr + (size_t)d * D_);
  const float4* xj = (const float4*)(xw + ((size_t)r * N_ + (size_t)s) * D_);
  const float4* ai = (const float4*)att;          // att[:D]
  const float4* aj = (const float4*)(att + D_);   // att[D:]
  float4 vi = xi[lane], vj = xj[lane], wi = ai[lane], wj = aj[lane];
  float p = vi.x * wi.x + vi.y * wi.y + vi.z * wi.z + vi.w * wi.w +
            vj.x * wj.x + vj.y * wj.y + vj.z * wj.z + vj.w * wj.w;
  p = wave_sum32(p);
  if (lane == 0) {
    float a = (p > 0.f) ? p : SLOPE_ * p;
    alpha[e] = a;
    atomicMax(amaxkey + d, f2key(a));
  }
}

// one thread per edge: ex = exp(alpha - amax[dst]); denom[dst] += ex (alpha buf reused for ex)
__global__ void edge_exp(float* __restrict__ alpha,
                         const unsigned* __restrict__ amaxkey,
                         const int* __restrict__ dst,
                         float* __restrict__ denom) {
  int e = blockIdx.x * blockDim.x + threadIdx.x;
  if (e >= E_) return;
  int d = dst[e];
  float amax = key2f(amaxkey[d]);
  float ex = __expf(alpha[e] - amax);
  alpha[e] = ex;
  atomic_add_f32(denom + d, ex);
}

// one wave32 per edge: aggr[dst] += x_j * (ex / (denom[dst] + 1e-16))
__global__ void edge_scatter(const float* __restrict__ xw,
                             const float* __restrict__ ex,
                             const float* __restrict__ denom,
                             const int* __restrict__ src,
                             const int* __restrict__ dst,
                             const int* __restrict__ et,
                             float* __restrict__ aggr) {
  int e = (blockIdx.x * blockDim.x + threadIdx.x) >> 5;
  int lane = threadIdx.x & 31;
  if (e >= E_) return;
  int s = src[e], d = dst[e], r = et[e];
  float a = ex[e] / (denom[d] + 1e-16f);
  const float4* xj = (const float4*)(xw + ((size_t)r * N_ + (size_t)s) * D_);
  float4 v = xj[lane];
  float* p = aggr + (size_t)d * D_ + lane * 4;
  atomic_add_f32(p + 0, v.x * a);
  atomic_add_f32(p + 1, v.y * a);
  atomic_add_f32(p + 2, v.z * a);
  atomic_add_f32(p + 3, v.w * a);
}

// one wave32 per node: out = tanh(LayerNorm(aggr + xr + bias))
__global__ void finalize(const float* __restrict__ aggr,
                         const float* __restrict__ xr,
                         const float* __restrict__ bias,
                         const float* __restrict__ lnw,
                         const float* __restrict__ lnb,
                         float* __restrict__ state_out,
                         float* __restrict__ final_out) {
  int node = (blockIdx.x * blockDim.x + threadIdx.x) >> 5;
  int lane = threadIdx.x & 31;
  if (node >= N_) return;
  float4 v = ((const float4*)(aggr + (size_t)node * D_))[lane];
  float4 u = ((const float4*)(xr + (size_t)node * D_))[lane];
  float4 b = ((const float4*)bias)[lane];
  float o[4] = {v.x + u.x + b.x, v.y + u.y + b.y, v.z + u.z + b.z, v.w + u.w + b.w};
  float s = wave_sum32(o[0] + o[1] + o[2] + o[3]);
  float mu = s * (1.0f / D_);
  float sq = 0.f;
#pragma unroll
  for (int j = 0; j < 4; ++j) { float dlt = o[j] - mu; sq += dlt * dlt; }
  sq = wave_sum32(sq);
  float inv = rsqrtf(sq * (1.0f / D_) + EPS_);
  float4 lw = ((const float4*)lnw)[lane];
  float4 lb = ((const float4*)lnb)[lane];
  float r0 = tanhf((o[0] - mu) * inv * lw.x + lb.x);
  float r1 = tanhf((o[1] - mu) * inv * lw.y + lb.y);
  float r2 = tanhf((o[2] - mu) * inv * lw.z + lb.z);
  float r3 = tanhf((o[3] - mu) * inv * lw.w + lb.w);
  float4 res = make_float4(r0, r1, r2, r3);
  ((float4*)(state_out + (size_t)node * D_))[lane] = res;
  if (final_out) ((float4*)(final_out + (size_t)node * D_))[lane] = res;
}

// ---------------- launcher ----------------
extern "C" void kernel_launch(void* const* d_in, const int* in_sizes, int n_in,
                              void* d_out, int out_size, void* d_ws, size_t ws_size,
                              hipStream_t stream) {
  const float* x0    = (const float*)d_in[0];
  const int*   ei    = (const int*)d_in[1];
  const int*   src   = ei;
  const int*   dst   = ei + E_;
  const int*   et    = (const int*)d_in[2];
  const float* basis = (const float*)d_in[3];
  const float* att_r = (const float*)d_in[4];
  const float* att   = (const float*)d_in[5];
  const float* root  = (const float*)d_in[6];
  const float* bias  = (const float*)d_in[7];
  const float* lnw   = (const float*)d_in[8];
  const float* lnb   = (const float*)d_in[9];

  float* out = (float*)d_out;  // [N*D final x] ++ [L*N*D stacked states]

  // workspace layout
  float*    w       = (float*)d_ws;                       // R*D*D
  float*    xw      = w + (size_t)R_ * D_ * D_;           // R*N*D
  float*    xr      = xw + (size_t)R_ * N_ * D_;          // N*D
  float*    alpha   = xr + (size_t)N_ * D_;               // E (alpha, then ex)
  unsigned* amaxkey = (unsigned*)(alpha + E_);            // N
  float*    denom   = (float*)(amaxkey + N_);             // N

  for (int l = 0; l < L_; ++l) {
    const float* xin = (l == 0) ? x0 : (out + (size_t)N_ * D_ + (size_t)(l - 1) * N_ * D_);
    float* aggr = out + (size_t)N_ * D_ + (size_t)l * N_ * D_;  // states[l] doubles as aggr buf

    make_w<<<(R_ * D_ * D_ + 255) / 256, 256, 0, stream>>>(
        att_r + l * R_ * B_, basis + (size_t)l * B_ * D_ * D_, w);

    gemm_xw<<<dim3(N_ / 16, 1, R_ + 1), 256, 0, stream>>>(
        xin, w, root + (size_t)l * D_ * D_, xw, xr);

    zero_bufs<<<(N_ * D_ + 255) / 256, 256, 0, stream>>>(aggr, amaxkey, denom);

    edge_alpha<<<(E_ * 32 + 255) / 256, 256, 0, stream>>>(
        xw, xr, att + l * 2 * D_, src, dst, et, alpha, amaxkey);

    edge_exp<<<(E_ + 255) / 256, 256, 0, stream>>>(alpha, amaxkey, dst, denom);

    edge_scatter<<<(E_ * 32 + 255) / 256, 256, 0, stream>>>(
        xw, alpha, denom, src, dst, et, aggr);

    finalize<<<(N_ * 32 + 255) / 256, 256, 0, stream>>>(
        aggr, xr, bias + l * D_, lnw + l * D_, lnb + l * D_, aggr,
        (l == L_ - 1) ? out : nullptr);
  }
}